// ManualGCN_32744830665359
// MI455X (gfx1250) — compile-verified
//
#include <hip/hip_runtime.h>

// ---------------------------------------------------------------------------
// 2-layer GCN for MI455X (gfx1250, wave32).
//   deg/norm -> [WMMA f32 GEMM h1=xW1] -> selfloop+bias -> edge scatter (HW f32
//   atomics) -> [WMMA f32 GEMM h2=relu(agg1)W2, ReLU fused] -> selfloop+bias ->
//   edge scatter into d_out.
// GEMMs use V_WMMA_F32_16X16X4_F32 (exact f32 path; GEMMs are memory-bound at
// AI~33 FLOP/B so f32 WMMA already saturates). The dominant cost is the edge
// scatter (~2 GB traffic, ~100us at 23.3 TB/s) done with global_atomic_add_f32.
// ---------------------------------------------------------------------------

typedef __attribute__((ext_vector_type(2))) float v2f;
typedef __attribute__((ext_vector_type(8))) float v8f;

#define N_NODES 100000
#define N_EDGES 1600000
#define IN_F    128
#define HID     128
#define OUT_F   64

// ---------------- degree / normalization ----------------------------------
__global__ void k_deg_init(float* __restrict__ deg) {
  int i = blockIdx.x * blockDim.x + threadIdx.x;
  if (i < N_NODES) deg[i] = 1.0f;   // self-loop contributes 1
}

__global__ void k_deg_accum(const int* __restrict__ dst, float* __restrict__ deg) {
  int e = blockIdx.x * blockDim.x + threadIdx.x;
  if (e < N_EDGES) unsafeAtomicAdd(&deg[dst[e]], 1.0f);  // global_atomic_add_f32
}

__global__ void k_rsqrt(float* __restrict__ dinv) {
  int i = blockIdx.x * blockDim.x + threadIdx.x;
  if (i < N_NODES) dinv[i] = rsqrtf(dinv[i]);
}

// ---------------- WMMA f32 GEMM: out[N,M] = (RELU?)A[N,128] * W[128,M] ------
// One wave owns a 16-row strip and all M/16 column tiles (A fragments reused
// across tiles). W staged in LDS (64KB/32KB << 320KB WGP LDS).
// A-fragment layout (ISA 7.12.2, 32-bit A 16x4): lanes 0-15 rows M=0..15 hold
// K={k,k+1} in v0/v1; lanes 16-31 hold K={k+2,k+3}.
// C/D layout: VGPR j -> row j (lanes 0-15) / row j+8 (lanes 16-31), col = lane&15.
template <int M, bool RELU>
__global__ void __launch_bounds__(256)
k_gemm(const float* __restrict__ A, const float* __restrict__ W,
       float* __restrict__ out, int nStrips) {
  constexpr int MT = M / 16;
  __shared__ float sW[128 * M];

  // cooperative LDS stage of W (row-major [128 x M])
  {
    const float4* Wg = (const float4*)W;
    float4* Ws = (float4*)sW;
    for (int i = threadIdx.x; i < (128 * M) / 4; i += 256) Ws[i] = Wg[i];
  }
  __syncthreads();

  int strip = blockIdx.x * 8 + (threadIdx.x >> 5);
  if (strip >= nStrips) return;            // wave-uniform; EXEC stays all-ones

  const int lane = threadIdx.x & 31;
  const int half = lane >> 4;              // 0: K pair {k,k+1}, 1: {k+2,k+3}
  const int l16  = lane & 15;

  const float* Ap = A + (size_t)(strip * 16 + l16) * 128;

  v8f acc[MT] = {};

  for (int k = 0; k < 128; k += 4) {
    const int kb = k + 2 * half;
    float ax = Ap[kb];
    float ay = Ap[kb + 1];
    if (RELU) { ax = fmaxf(ax, 0.0f); ay = fmaxf(ay, 0.0f); }
    v2f a; a.x = ax; a.y = ay;
#pragma unroll
    for (int t = 0; t < MT; ++t) {
      const int col = t * 16 + l16;
      v2f b; b.x = sW[kb * M + col]; b.y = sW[(kb + 1) * M + col];
      acc[t] = __builtin_amdgcn_wmma_f32_16x16x4_f32(
          /*neg_a=*/false, a, /*neg_b=*/false, b,
          /*c_mod=*/(short)0, acc[t], /*reuse_a=*/false, /*reuse_b=*/false);
    }
  }

#pragma unroll
  for (int t = 0; t < MT; ++t) {
    const int col = t * 16 + l16;
#pragma unroll
    for (int j = 0; j < 8; ++j) {
      const size_t row = (size_t)(strip * 16 + j + 8 * half);
      out[row * M + col] = acc[t][j];
    }
  }
}

// ---------------- self-loop + bias fold: out = b + dinv^2 * h ---------------
template <int D>
__global__ void k_init_out(const float* __restrict__ h, const float* __restrict__ dinv,
                           const float* __restrict__ bias, float* __restrict__ out) {
  constexpr int DV = D / 4;
  int idx = blockIdx.x * blockDim.x + threadIdx.x;   // over N_NODES * DV float4s
  if (idx >= N_NODES * DV) return;
  const int i  = idx / DV;
  const int f4 = idx - i * DV;
  const float di = dinv[i];
  const float s = di * di;
  float4 hv = ((const float4*)(h + (size_t)i * D))[f4];
  float4 bv = ((const float4*)bias)[f4];
  float4 o;
  o.x = bv.x + s * hv.x;
  o.y = bv.y + s * hv.y;
  o.z = bv.z + s * hv.z;
  o.w = bv.w + s * hv.w;
  ((float4*)(out + (size_t)i * D))[f4] = o;
}

// ---------------- edge scatter: out[dst] += dinv[s]*dinv[d] * h[src] --------
// One wave per edge; float4 (D=128) / float2 (D=64) per lane; HW f32 atomics.
template <int D>
__global__ void __launch_bounds__(256)
k_scatter(const float* __restrict__ h, const float* __restrict__ dinv,
          const int* __restrict__ srcA, const int* __restrict__ dstA,
          float* __restrict__ out) {
  int wid = blockIdx.x * (blockDim.x >> 5) + (threadIdx.x >> 5);
  if (wid >= N_EDGES) return;
  const int lane = threadIdx.x & 31;
  const int s = srcA[wid];
  const int d = dstA[wid];
  const float nrm = dinv[s] * dinv[d];
  if (D == 128) {
    float4 v = ((const float4*)(h + (size_t)s * 128))[lane];   // global_load_b128
    float* o = out + (size_t)d * 128 + lane * 4;
    unsafeAtomicAdd(o + 0, v.x * nrm);
    unsafeAtomicAdd(o + 1, v.y * nrm);
    unsafeAtomicAdd(o + 2, v.z * nrm);
    unsafeAtomicAdd(o + 3, v.w * nrm);
  } else {
    float2 v = ((const float2*)(h + (size_t)s * 64))[lane];
    float* o = out + (size_t)d * 64 + lane * 2;
    unsafeAtomicAdd(o + 0, v.x * nrm);
    unsafeAtomicAdd(o + 1, v.y * nrm);
  }
}

// ---------------------------------------------------------------------------
extern "C" void kernel_launch(void* const* d_in, const int* in_sizes, int n_in,
                              void* d_out, int out_size, void* d_ws, size_t ws_size,
                              hipStream_t stream) {
  const float* x  = (const float*)d_in[0];   // [N, 128]
  const int*   ei = (const int*)d_in[1];     // [2, E]
  const float* W1 = (const float*)d_in[2];   // [128, 128]
  const float* b1 = (const float*)d_in[3];   // [128]
  const float* W2 = (const float*)d_in[4];   // [128, 64]
  const float* b2 = (const float*)d_in[5];   // [64]
  float* out = (float*)d_out;                // [N, 64]

  const int* srcA = ei;                      // edge_index[0]
  const int* dstA = ei + N_EDGES;            // edge_index[1]

  // workspace layout (~129 MB): dinv | h1 | agg1 | h2
  char* ws = (char*)d_ws;
  float* dinv = (float*)ws;                                 // N floats
  float* h1   = (float*)(ws + (1 << 19));                   // N*128
  float* agg1 = h1 + (size_t)N_NODES * 128;                 // N*128
  float* h2   = agg1 + (size_t)N_NODES * 128;               // N*64

  const int nStrips = N_NODES / 16;   // 6250, exact

  // normalization
  k_deg_init <<<(N_NODES + 255) / 256, 256, 0, stream>>>(dinv);
  k_deg_accum<<<(N_EDGES + 255) / 256, 256, 0, stream>>>(dstA, dinv);
  k_rsqrt    <<<(N_NODES + 255) / 256, 256, 0, stream>>>(dinv);

  // layer 1
  k_gemm<128, false><<<(nStrips + 7) / 8, 256, 0, stream>>>(x, W1, h1, nStrips);
  k_init_out<128>   <<<(N_NODES * 32 + 255) / 256, 256, 0, stream>>>(h1, dinv, b1, agg1);
  k_scatter<128>    <<<(N_EDGES + 7) / 8, 256, 0, stream>>>(h1, dinv, srcA, dstA, agg1);

  // layer 2 (ReLU fused into GEMM A-load)
  k_gemm<64, true><<<(nStrips + 7) / 8, 256, 0, stream>>>(agg1, W2, h2, nStrips);
  k_init_out<64>  <<<(N_NODES * 16 + 255) / 256, 256, 0, stream>>>(h2, dinv, b2, out);
  k_scatter<64>   <<<(N_EDGES + 7) / 8, 256, 0, stream>>>(h2, dinv, srcA, dstA, out);
}